// SSIMStabilityLoss_13116830122520
// MI455X (gfx1250) — compile-verified
//
#include <hip/hip_runtime.h>

typedef __attribute__((ext_vector_type(2))) float v2f;
typedef __attribute__((ext_vector_type(8))) float v8f;

// D = A(16x4,f32) x B(4x16,f32) + C(16x16,f32), full fp32 matrix op on CDNA5.
#define WMMA4(a, b, c) \
  __builtin_amdgcn_wmma_f32_16x16x4_f32(false, (a), false, (b), (short)0, (c), false, false)

namespace {

constexpr int H = 512, W = 512;
constexpr int TILE = 16;
constexpr int TPR = W / TILE;        // 32 tiles per row
constexpr int TPI = (H / TILE) * TPR;// 1024 tiles per image
constexpr int NIMG = 32;
constexpr int NTILES = NIMG * TPI;   // 32768
constexpr int WAVES = 2;             // tiles (waves) per workgroup -> 64 threads
constexpr int PS = 40;               // x/y staging plane row stride (floats)
constexpr int PROWS = 28;            // K padded to 28 (7 chunks of 4)
constexpr int VS = 40;               // V plane row stride
constexpr int PLANE = PROWS * PS;    // 1120 floats
constexpr int VPLANE = TILE * VS;    // 640 floats
constexpr int WAVE_LDS = 2 * PLANE + 5 * VPLANE; // 5440 floats = 21.25 KB/wave
constexpr float C1 = 0.0001f;        // (0.01*1.0)^2
constexpr float C2 = 0.0009f;        // (0.03*1.0)^2

// 11-tap Gaussian, sigma=1.5, normalized (matches reference to fp32)
__device__ __constant__ float GW[11] = {
    0.00102838f, 0.00759880f, 0.03600077f, 0.10936063f, 0.21300553f,
    0.26601172f,
    0.21300553f, 0.10936063f, 0.03600077f, 0.00759880f, 0.00102838f};

__global__ void zero_acc_kernel(double* acc) {
  if (threadIdx.x == 0 && blockIdx.x == 0) *acc = 0.0;
}

__global__ __launch_bounds__(WAVES * 32) void ssim_wmma_kernel(
    const float* __restrict__ X, const float* __restrict__ Y,
    double* __restrict__ acc) {
  __shared__ float lds[WAVES * WAVE_LDS];

  const int lane = threadIdx.x & 31;
  const int wv = threadIdx.x >> 5;
  float* xs = lds + wv * WAVE_LDS;
  float* ys = xs + PLANE;
  float* vp = ys + PLANE;  // 5 planes of 16 x VS (vertical-conv scratch)

  const int T = blockIdx.x * WAVES + wv;  // one 16x16 output tile per wave
  const int img = T / TPI;
  const int rem = T - img * TPI;
  const int ty = rem >> 5;            // TPR == 32
  const int tx = rem & (TPR - 1);
  const int row0 = ty * TILE, col0 = tx * TILE;
  const float* Xi = X + (size_t)img * (H * W);
  const float* Yi = Y + (size_t)img * (H * W);

  // Zero both staging planes (covers SAME zero-padding + K/col padding).
  for (int i = lane; i < 2 * PLANE; i += 32) xs[i] = 0.0f;

  // Stage the 26x26 halo regions (rows/cols of L2-resident input; halo
  // overfetch is served by the 192MB L2, HBM sees ~unique bytes only).
  for (int i = lane; i < 26 * 26; i += 32) {
    const int rr = i / 26, cc = i - rr * 26;
    const int gr = row0 - 5 + rr, gc = col0 - 5 + cc;
    if ((unsigned)gr < (unsigned)H && (unsigned)gc < (unsigned)W) {
      xs[rr * PS + cc] = Xi[gr * W + gc];
      ys[rr * PS + cc] = Yi[gr * W + gc];
    }
  }
  // Single wave per slab: same-wave LDS ops are ordered; compiler emits
  // the s_wait_dscnt it needs before dependent reads.

  const int r = lane & 15;
  const int hi = lane >> 4;
  const int c0 = r, c1 = r + 16;

  // Banded Gaussian fragment values g[k - r], k = 4t + 2hi + v.
  // Serves as the A fragment of the vertical pass (Wv[r][k]) AND the
  // B fragment of the horizontal pass (Wh[k][r]) -- computed once.
  v2f wk[7];
#pragma unroll
  for (int t = 0; t < 7; ++t) {
    const int k0 = 4 * t + 2 * hi;
    const int d0 = k0 - r, d1 = k0 + 1 - r;
    v2f w;
    w.x = ((unsigned)d0 < 11u) ? GW[d0] : 0.0f;
    w.y = ((unsigned)d1 < 11u) ? GW[d1] : 0.0f;
    wk[t] = w;
  }

  // ---- Vertical pass: V_p(16x32) = Wv(16x28) x P_p(28x32)
  // Chunk-outer so each LDS read of x/y feeds all 10 accumulators
  // (5 maps {x, y, xx, yy, xy} x 2 column tiles).
  v8f a00 = {}, a01 = {}, a10 = {}, a11 = {}, a20 = {}, a21 = {}, a30 = {},
      a31 = {}, a40 = {}, a41 = {};
#pragma unroll
  for (int t = 0; t < 7; ++t) {
    const int k0 = 4 * t + 2 * hi;
    const float xa0 = xs[k0 * PS + c0], xb0 = xs[(k0 + 1) * PS + c0];
    const float xa1 = xs[k0 * PS + c1], xb1 = xs[(k0 + 1) * PS + c1];
    const float ya0 = ys[k0 * PS + c0], yb0 = ys[(k0 + 1) * PS + c0];
    const float ya1 = ys[k0 * PS + c1], yb1 = ys[(k0 + 1) * PS + c1];
    const v2f a = wk[t];
    v2f b;
    b.x = xa0;       b.y = xb0;       a00 = WMMA4(a, b, a00);
    b.x = xa1;       b.y = xb1;       a01 = WMMA4(a, b, a01);
    b.x = ya0;       b.y = yb0;       a10 = WMMA4(a, b, a10);
    b.x = ya1;       b.y = yb1;       a11 = WMMA4(a, b, a11);
    b.x = xa0 * xa0; b.y = xb0 * xb0; a20 = WMMA4(a, b, a20);
    b.x = xa1 * xa1; b.y = xb1 * xb1; a21 = WMMA4(a, b, a21);
    b.x = ya0 * ya0; b.y = yb0 * yb0; a30 = WMMA4(a, b, a30);
    b.x = ya1 * ya1; b.y = yb1 * yb1; a31 = WMMA4(a, b, a31);
    b.x = xa0 * ya0; b.y = xb0 * yb0; a40 = WMMA4(a, b, a40);
    b.x = xa1 * ya1; b.y = xb1 * yb1; a41 = WMMA4(a, b, a41);
  }

  // Spill V planes: D element v lives at row (v + 8*hi), col (lane&15).
#pragma unroll
  for (int v = 0; v < 8; ++v) {
    const int m = (v + 8 * hi) * VS;
    vp[0 * VPLANE + m + c0] = a00[v]; vp[0 * VPLANE + m + c1] = a01[v];
    vp[1 * VPLANE + m + c0] = a10[v]; vp[1 * VPLANE + m + c1] = a11[v];
    vp[2 * VPLANE + m + c0] = a20[v]; vp[2 * VPLANE + m + c1] = a21[v];
    vp[3 * VPLANE + m + c0] = a30[v]; vp[3 * VPLANE + m + c1] = a31[v];
    vp[4 * VPLANE + m + c0] = a40[v]; vp[4 * VPLANE + m + c1] = a41[v];
  }

  // ---- Horizontal pass: Out_p(16x16) = V_p(16x28) x Wh(28x16)
  v8f o0 = {}, o1 = {}, o2 = {}, o3 = {}, o4 = {};
#pragma unroll
  for (int t = 0; t < 7; ++t) {
    const int k0 = 4 * t + 2 * hi;
    const v2f b = wk[t];
    const int rb = r * VS + k0;
    v2f a;
    a.x = vp[0 * VPLANE + rb]; a.y = vp[0 * VPLANE + rb + 1]; o0 = WMMA4(a, b, o0);
    a.x = vp[1 * VPLANE + rb]; a.y = vp[1 * VPLANE + rb + 1]; o1 = WMMA4(a, b, o1);
    a.x = vp[2 * VPLANE + rb]; a.y = vp[2 * VPLANE + rb + 1]; o2 = WMMA4(a, b, o2);
    a.x = vp[3 * VPLANE + rb]; a.y = vp[3 * VPLANE + rb + 1]; o3 = WMMA4(a, b, o3);
    a.x = vp[4 * VPLANE + rb]; a.y = vp[4 * VPLANE + rb + 1]; o4 = WMMA4(a, b, o4);
  }

  // ---- SSIM map + reduction
  float s = 0.0f;
#pragma unroll
  for (int v = 0; v < 8; ++v) {
    const float mx = o0[v], my = o1[v];
    const float mx2 = mx * mx, my2 = my * my, mxy = mx * my;
    const float sx = o2[v] - mx2;
    const float sy = o3[v] - my2;
    const float sxy = o4[v] - mxy;
    const float num = (2.0f * mxy + C1) * (2.0f * sxy + C2);
    const float den = (mx2 + my2 + C1) * (sx + sy + C2);
    s += num / den;
  }
#pragma unroll
  for (int off = 16; off > 0; off >>= 1) s += __shfl_xor(s, off, 32);
  if (lane == 0) atomicAdd(acc, (double)s);  // GLOBAL_ATOMIC_ADD_F64
}

__global__ void finalize_kernel(const double* __restrict__ acc,
                                float* __restrict__ out) {
  if (threadIdx.x == 0 && blockIdx.x == 0) {
    out[0] = 1.0f - (float)(*acc / (double)((long long)NIMG * H * W));
  }
}

}  // namespace

extern "C" void kernel_launch(void* const* d_in, const int* in_sizes, int n_in,
                              void* d_out, int out_size, void* d_ws,
                              size_t ws_size, hipStream_t stream) {
  (void)in_sizes; (void)n_in; (void)out_size; (void)ws_size;
  const float* X = (const float*)d_in[0];  // heatmap_clean
  const float* Y = (const float*)d_in[1];  // heatmap_adv
  float* out = (float*)d_out;
  double* acc = (double*)d_ws;  // 8 bytes of scratch, re-zeroed every call

  zero_acc_kernel<<<1, 1, 0, stream>>>(acc);
  ssim_wmma_kernel<<<NTILES / WAVES, WAVES * 32, 0, stream>>>(X, Y, acc);
  finalize_kernel<<<1, 1, 0, stream>>>(acc, out);
}